// AttentionAggregator_75677323756077
// MI455X (gfx1250) — compile-verified
//
#include <hip/hip_runtime.h>
#include <hip/hip_bf16.h>

#define IN_DIM 256
#define OUT_DIM 128
#define SLOPE 0.1f

typedef float v2f __attribute__((ext_vector_type(2)));
typedef float v8f __attribute__((ext_vector_type(8)));

// ---------------------------------------------------------------------------
// Kernel 1: h[row, :] = features[unique_nodes[row], :] @ W^T + b
// One block = 256 threads = 8 waves. Block handles a 16-row tile; wave w
// handles output columns [16w, 16w+16). Uses V_WMMA_F32_16X16X4_F32.
// A layout (16x4 f32): lane m=lane&15, K-pair = lane>>4 -> contiguous float2.
// B layout (4x16 f32): lane n=lane&15, K-pair = lane>>4 -> contiguous float2
//   from W[n, :] (B = W^T means B[k][n] = W[n][k]).
// C/D layout (16x16 f32): lane n=lane&15, vgpr r -> m = r + 8*(lane>>4).
// ---------------------------------------------------------------------------
__global__ __launch_bounds__(256) void gat_gemm_kernel(
    const float* __restrict__ features, const float* __restrict__ W,
    const float* __restrict__ bias, const int* __restrict__ unique_nodes,
    float* __restrict__ h, int N)
{
    const int lane = threadIdx.x & 31;
    const int wave = threadIdx.x >> 5;      // 0..7 : column tile
    const int half = lane >> 4;             // K-pair select
    const int mn   = lane & 15;             // m for A, n for B/C/D

    const int rowBase = blockIdx.x * 16;
    int arow_idx = rowBase + mn;
    if (arow_idx >= N) arow_idx = N - 1;           // clamp (N%16==0 normally)
    const int gsrc = unique_nodes[arow_idx];
    const float* __restrict__ arow = features + (size_t)gsrc * IN_DIM;

    const int ncol = wave * 16 + mn;
    const float* __restrict__ brow = W + (size_t)ncol * IN_DIM;

    // Fold bias into the accumulator: every vgpr r of this lane is column ncol.
    v8f acc;
    const float bv = bias[ncol];
#pragma unroll
    for (int r = 0; r < 8; ++r) acc[r] = bv;

#pragma unroll 4
    for (int k = 0; k < IN_DIM; k += 4) {
        v2f av  = *(const v2f*)(arow + k + 2 * half);
        v2f bvv = *(const v2f*)(brow + k + 2 * half);
        acc = __builtin_amdgcn_wmma_f32_16x16x4_f32(
            /*neg_a=*/false, av, /*neg_b=*/false, bvv,
            /*c_mod=*/(short)0, acc, /*reuse_a=*/false, /*reuse_b=*/false);
    }

    // Store: this lane's 8 outputs live at rows rowBase+8*half+r, column ncol.
    float* __restrict__ obase =
        h + (size_t)(rowBase + 8 * half) * OUT_DIM + ncol;
    if (rowBase + 16 <= N) {
        // full tile (always true when N % 16 == 0): unconditional stores with
        // immediate offsets r*OUT_DIM*4 bytes — no per-store exec masking.
#pragma unroll
        for (int r = 0; r < 8; ++r)
            obase[(size_t)r * OUT_DIM] = acc[r];
    } else {
#pragma unroll
        for (int r = 0; r < 8; ++r) {
            int orow = rowBase + 8 * half + r;
            if (orow < N) obase[(size_t)r * OUT_DIM] = acc[r];
        }
    }
}

// ---------------------------------------------------------------------------
// Kernel 2: s_src[i] = h[i,:] . a[0:128], s_dst[i] = h[i,:] . a[128:256]
// One wave per row; lane covers 4 columns via float4; wave32 xor-reduction.
// ---------------------------------------------------------------------------
__global__ __launch_bounds__(256) void gat_scores_kernel(
    const float* __restrict__ h, const float* __restrict__ a,
    float* __restrict__ s_src, float* __restrict__ s_dst, int N)
{
    const int lane = threadIdx.x & 31;
    const int row  = blockIdx.x * 8 + (threadIdx.x >> 5);
    if (row >= N) return;

    const float4 hv = *(const float4*)(h + (size_t)row * OUT_DIM + lane * 4);
    const float4 as = *(const float4*)(a + lane * 4);
    const float4 ad = *(const float4*)(a + OUT_DIM + lane * 4);

    float ps = hv.x * as.x + hv.y * as.y + hv.z * as.z + hv.w * as.w;
    float pd = hv.x * ad.x + hv.y * ad.y + hv.z * ad.z + hv.w * ad.w;

#pragma unroll
    for (int off = 16; off > 0; off >>= 1) {
        ps += __shfl_xor(ps, off, 32);
        pd += __shfl_xor(pd, off, 32);
    }
    if (lane == 0) {
        s_src[row] = ps;
        s_dst[row] = pd;
    }
}

// ---------------------------------------------------------------------------
// Kernel 3: per-edge scatter. One wave per edge; lane handles 4 channels.
// num[src,:] += w * h[dst,:]; den[src] += w;  w = exp(leaky_relu(s))
// Uses HW global_atomic_add_f32 via unsafeAtomicAdd.
// ---------------------------------------------------------------------------
__global__ __launch_bounds__(256) void gat_edge_kernel(
    const int* __restrict__ edges, const float* __restrict__ h,
    const float* __restrict__ s_src, const float* __restrict__ s_dst,
    float* __restrict__ num, float* __restrict__ den, int E)
{
    const int lane = threadIdx.x & 31;
    const int e    = blockIdx.x * 8 + (threadIdx.x >> 5);
    if (e >= E) return;

    const int src = edges[2 * e + 0];
    const int dst = edges[2 * e + 1];

    float s = s_src[src] + s_dst[dst];
    s = (s >= 0.0f) ? s : SLOPE * s;
    const float w = expf(s);

    const float4 hv = *(const float4*)(h + (size_t)dst * OUT_DIM + lane * 4);
    float* np = num + (size_t)src * OUT_DIM + lane * 4;
    unsafeAtomicAdd(np + 0, w * hv.x);
    unsafeAtomicAdd(np + 1, w * hv.y);
    unsafeAtomicAdd(np + 2, w * hv.z);
    unsafeAtomicAdd(np + 3, w * hv.w);
    if (lane == 0) unsafeAtomicAdd(&den[src], w);
}

// ---------------------------------------------------------------------------
// Kernel 4: out[j,:] = num[node_idx[j],:] / den[node_idx[j]]
// One wave per output row; lane covers 4 channels.
// ---------------------------------------------------------------------------
__global__ __launch_bounds__(256) void gat_out_kernel(
    const float* __restrict__ num, const float* __restrict__ den,
    const int* __restrict__ node_idx, float* __restrict__ out, int B)
{
    const int t = blockIdx.x * blockDim.x + threadIdx.x;
    const int j = t >> 5;
    const int lane = t & 31;
    if (j >= B) return;

    const int row = node_idx[j];
    const float inv = 1.0f / den[row];
    const float4 nv = *(const float4*)(num + (size_t)row * OUT_DIM + lane * 4);
    float4 ov;
    ov.x = nv.x * inv; ov.y = nv.y * inv; ov.z = nv.z * inv; ov.w = nv.w * inv;
    *(float4*)(out + (size_t)j * OUT_DIM + lane * 4) = ov;
}

extern "C" void kernel_launch(void* const* d_in, const int* in_sizes, int n_in,
                              void* d_out, int out_size, void* d_ws, size_t ws_size,
                              hipStream_t stream) {
    const float* features     = (const float*)d_in[0];   // NODE_NUM x 256
    const float* W            = (const float*)d_in[1];   // 128 x 256
    const float* b            = (const float*)d_in[2];   // 128
    const float* a            = (const float*)d_in[3];   // 256 x 1
    const int*   edges        = (const int*)d_in[4];     // E x 2
    const int*   unique_nodes = (const int*)d_in[5];     // N
    const int*   node_idx     = (const int*)d_in[6];     // B
    float* out = (float*)d_out;

    const int N = in_sizes[5];
    const int E = in_sizes[4] / 2;
    const int B = in_sizes[6];

    // workspace layout
    float* h     = (float*)d_ws;                         // N * 128
    float* num   = h + (size_t)N * OUT_DIM;              // N * 128
    float* s_src = num + (size_t)N * OUT_DIM;            // N
    float* s_dst = s_src + N;                            // N
    float* den   = s_dst + N;                            // N

    hipMemsetAsync(num, 0, (size_t)N * OUT_DIM * sizeof(float), stream);
    hipMemsetAsync(den, 0, (size_t)N * sizeof(float), stream);

    // 1) WMMA GEMM: h = features[unique_nodes] @ W^T + b
    {
        dim3 grid((N + 15) / 16);
        gat_gemm_kernel<<<grid, 256, 0, stream>>>(features, W, b, unique_nodes, h, N);
    }
    // 2) attention score vectors
    {
        dim3 grid((N + 7) / 8);
        gat_scores_kernel<<<grid, 256, 0, stream>>>(h, a, s_src, s_dst, N);
    }
    // 3) edge scatter-accumulate
    {
        dim3 grid((E + 7) / 8);
        gat_edge_kernel<<<grid, 256, 0, stream>>>(edges, h, s_src, s_dst, num, den, E);
    }
    // 4) normalize + gather output rows
    {
        long threads = (long)B * 32;
        dim3 grid((unsigned)((threads + 255) / 256));
        gat_out_kernel<<<grid, 256, 0, stream>>>(num, den, node_idx, out, B);
    }
}